// RWKV_Tmix_18648747999313
// MI455X (gfx1250) — compile-verified
//
#include <hip/hip_runtime.h>
#include <math.h>

// RWKV6 time-mix for MI455X (gfx1250, wave32).
// B=32, T=2048, C=64, H=1, N=64.  All math fp32; dense GEMMs on
// V_WMMA_F32_16X16X4_F32; the sequential WKV scan keeps the 64x64 state in
// VGPRs (one column per lane), broadcasting r/k/dec from LDS.

#define Bb 32
#define Tt 2048
#define Cc 64
#define Mrows (Bb * Tt)          // 65536 rows
#define LDA 66                   // padded LDS row stride (floats)
#define LDAA 162                 // padded stride for the 16x160 'a' tile
#define LN_EPS 6.4e-4f           // 1e-5 * 8^2

typedef __attribute__((ext_vector_type(2))) float v2f;
typedef __attribute__((ext_vector_type(8))) float v8f;

// ---- WMMA helpers (fp32 16x16x4, wave32) -----------------------------------
// A: 16xK tile in LDS, row-major stride sA. Fragment: lane l -> m=l%16,
//    h=l/16, element q holds A[m][k0 + 2h + q].
// B (non-transposed): B[k][n] = Bp[k*sB + n]  (weight used as-is)
// B (transposed):     B[k][n] = Wp[n*sW + k]  (weight used as W^T; caller
//    pre-offsets Wp by the output-column tile: Wp += ncol0*sW)

__device__ __forceinline__ v8f wmma_AB(const float* A, int sA,
                                       const float* Bp, int sB,
                                       int K, v8f acc) {
  const int l = threadIdx.x & 31;
  const int mn = l & 15;
  const int h  = l >> 4;
  for (int k0 = 0; k0 < K; k0 += 4) {
    const int kb = k0 + 2 * h;
    v2f a, b;
    a.x = A[mn * sA + kb];
    a.y = A[mn * sA + kb + 1];
    b.x = Bp[kb * sB + mn];
    b.y = Bp[(kb + 1) * sB + mn];
    acc = __builtin_amdgcn_wmma_f32_16x16x4_f32(false, a, false, b,
                                                (short)0, acc, false, false);
  }
  return acc;
}

__device__ __forceinline__ v8f wmma_ABt(const float* A, int sA,
                                        const float* Wp, int sW,
                                        int K, v8f acc) {
  const int l = threadIdx.x & 31;
  const int mn = l & 15;
  const int h  = l >> 4;
  for (int k0 = 0; k0 < K; k0 += 4) {
    const int kb = k0 + 2 * h;
    v2f a, b;
    a.x = A[mn * sA + kb];
    a.y = A[mn * sA + kb + 1];
    b.x = Wp[mn * sW + kb];
    b.y = Wp[mn * sW + kb + 1];
    acc = __builtin_amdgcn_wmma_f32_16x16x4_f32(false, a, false, b,
                                                (short)0, acc, false, false);
  }
  return acc;
}

// ---- Kernel 1: token-shift mixing + all input projections ------------------
// Per block: one 16-row M tile, 128 threads (4 waves).  Produces
// r, k, v, silu(g), tanh(xw @ dec_w1) for those rows.
__global__ __launch_bounds__(128) void k_mix_proj(
    const float* __restrict__ x,
    const float* __restrict__ tmx, const float* __restrict__ tmw,
    const float* __restrict__ tmk, const float* __restrict__ tmv,
    const float* __restrict__ tmr, const float* __restrict__ tmg,
    const float* __restrict__ maa_w1, const float* __restrict__ maa_w2,
    const float* __restrict__ dec_w1,
    const float* __restrict__ Wr, const float* __restrict__ Wk,
    const float* __restrict__ Wv, const float* __restrict__ Wg,
    float* __restrict__ Rb, float* __restrict__ Kb, float* __restrict__ Vb,
    float* __restrict__ Gb, float* __restrict__ Hb)
{
  __shared__ float sX[16 * LDA];
  __shared__ float sXX[16 * LDA];
  __shared__ float sXXX[16 * LDA];
  __shared__ float sA[16 * LDAA];          // 16 x 160 tanh(xxx@maa_w1)
  __shared__ float sXf[5][16 * LDA];       // xw, xk, xv, xr, xg

  const int row0 = blockIdx.x * 16;
  const int tid  = threadIdx.x;
  const int wave = tid >> 5;

  // Stage x / xx / xxx tiles (token shift; t==0 -> prev row is 0)
  for (int i = tid; i < 16 * 64; i += 128) {
    const int r = i >> 6, c = i & 63;
    const int gm = row0 + r;
    const float xv_ = x[gm * 64 + c];
    const float xp  = ((gm & (Tt - 1)) == 0) ? 0.f : x[(gm - 1) * 64 + c];
    const float xx  = xp - xv_;
    sX[r * LDA + c]   = xv_;
    sXX[r * LDA + c]  = xx;
    sXXX[r * LDA + c] = xv_ + xx * tmx[c];
  }
  __syncthreads();

  // a = tanh(xxx @ maa_w1)  -> 16x160 (10 column tiles)
  for (int j = wave; j < 10; j += 4) {
    v8f acc = {};
    acc = wmma_AB(sXXX, LDA, maa_w1 + j * 16, 160, 64, acc);
    const int l = tid & 31, n = l & 15, h = l >> 4;
#pragma unroll
    for (int v = 0; v < 8; v++)
      sA[(v + 8 * h) * LDAA + j * 16 + n] = tanhf(acc[v]);
  }
  __syncthreads();

  // m_f = a[:, f*32:(f+1)*32] @ maa_w2[f]  ->  x_f = x + xx*(tm_f + m_f)
  for (int j = wave; j < 20; j += 4) {
    const int f = j >> 2, nt = j & 3;
    v8f acc = {};
    acc = wmma_AB(sA + f * 32, LDAA, maa_w2 + f * 2048 + nt * 16, 64, 32, acc);
    const float* tm = (f == 0) ? tmw : (f == 1) ? tmk : (f == 2) ? tmv
                     : (f == 3) ? tmr : tmg;
    const int l = tid & 31, n = l & 15, h = l >> 4;
#pragma unroll
    for (int v = 0; v < 8; v++) {
      const int r = v + 8 * h, c = nt * 16 + n;
      sXf[f][r * LDA + c] =
          sX[r * LDA + c] + sXX[r * LDA + c] * (tm[c] + acc[v]);
    }
  }
  __syncthreads();

  // Projections: r=xr@Wr^T, k=xk@Wk^T, v=xv@Wv^T, g=silu(xg@Wg^T),
  //              h=tanh(xw@dec_w1)
  for (int j = wave; j < 20; j += 4) {
    const int which = j >> 2, nt = j & 3;
    v8f acc = {};
    float* out;
    if (which == 0)      { acc = wmma_ABt(sXf[3], LDA, Wr + nt * 16 * 64, 64, 64, acc); out = Rb; }
    else if (which == 1) { acc = wmma_ABt(sXf[1], LDA, Wk + nt * 16 * 64, 64, 64, acc); out = Kb; }
    else if (which == 2) { acc = wmma_ABt(sXf[2], LDA, Wv + nt * 16 * 64, 64, 64, acc); out = Vb; }
    else if (which == 3) { acc = wmma_ABt(sXf[4], LDA, Wg + nt * 16 * 64, 64, 64, acc); out = Gb; }
    else                 { acc = wmma_AB (sXf[0], LDA, dec_w1 + nt * 16,  64, 64, acc); out = Hb; }
    const int l = tid & 31, n = l & 15, h = l >> 4;
#pragma unroll
    for (int v = 0; v < 8; v++) {
      float o = acc[v];
      if (which == 3)      o = o / (1.f + expf(-o));   // silu
      else if (which == 4) o = tanhf(o);
      out[(size_t)(row0 + v + 8 * h) * 64 + nt * 16 + n] = o;
    }
  }
}

// ---- Kernel 2: dec = exp(-exp(t_decay + h @ dec_w2)), in place over Hb -----
__global__ __launch_bounds__(128) void k_decay(
    const float* __restrict__ dec_w2, const float* __restrict__ t_decay,
    float* __restrict__ Hb)
{
  __shared__ float sH[16 * LDA];
  const int row0 = blockIdx.x * 16;
  const int tid  = threadIdx.x;
  const int wave = tid >> 5;
  for (int i = tid; i < 16 * 64; i += 128) {
    const int r = i >> 6, c = i & 63;
    sH[r * LDA + c] = Hb[(size_t)(row0 + r) * 64 + c];
  }
  __syncthreads();
  const int nt = wave;                      // 4 waves -> 4 column tiles
  v8f acc = {};
  acc = wmma_AB(sH, LDA, dec_w2 + nt * 16, 64, 64, acc);
  const int l = tid & 31, n = l & 15, h = l >> 4;
#pragma unroll
  for (int v = 0; v < 8; v++) {
    const int c = nt * 16 + n;
    const float w = t_decay[c] + acc[v];
    Hb[(size_t)(row0 + v + 8 * h) * 64 + c] = expf(-expf(w));
  }
}

// ---- Kernel 3: WKV6 scan. One block per batch, 64 threads; lane j owns
// state column S[:,j] in VGPRs. Chunked LDS staging + global prefetch. ------
__global__ __launch_bounds__(64) void k_wkv(
    const float* __restrict__ Rb, const float* __restrict__ Kb,
    const float* __restrict__ Vb, const float* __restrict__ Db,
    const float* __restrict__ u,
    float* __restrict__ Yb)
{
  const int b = blockIdx.x;
  const int j = threadIdx.x;                // 0..63
  __shared__ float lr[16 * 64], lk[16 * 64], luk[16 * 64],
                   lv[16 * 64], ld[16 * 64], su[64];
  su[j] = u[j];
  float S[64];
#pragma unroll
  for (int i = 0; i < 64; i++) S[i] = 0.f;
  __syncthreads();

  for (int tc = 0; tc < Tt; tc += 16) {
    const size_t base = (size_t)(b * Tt + tc) * 64;
    for (int s = 0; s < 16; s++) {
      const float kk = Kb[base + s * 64 + j];
      lr[s * 64 + j]  = Rb[base + s * 64 + j];
      lk[s * 64 + j]  = kk;
      luk[s * 64 + j] = kk * su[j];
      lv[s * 64 + j]  = Vb[base + s * 64 + j];
      ld[s * 64 + j]  = Db[base + s * 64 + j];
    }
    if (tc + 16 < Tt) {                       // pull next chunk toward L2/L0
      __builtin_prefetch(&Rb[base + 1024 + j], 0, 1);
      __builtin_prefetch(&Kb[base + 1024 + j], 0, 1);
      __builtin_prefetch(&Vb[base + 1024 + j], 0, 1);
      __builtin_prefetch(&Db[base + 1024 + j], 0, 1);
    }
    __syncthreads();
    for (int s = 0; s < 16; s++) {
      const float vj = lv[s * 64 + j];
      float acc = 0.f, cc = 0.f;
#pragma unroll
      for (int i = 0; i < 64; i++) {
        const float ri = lr[s * 64 + i];
        acc += ri * S[i];
        cc  += ri * luk[s * 64 + i];
        S[i] = S[i] * ld[s * 64 + i] + lk[s * 64 + i] * vj;
      }
      Yb[base + s * 64 + j] = acc + cc * vj;  // y = r.(S) + (r.u*k) * v_j
    }
    __syncthreads();
  }
}

// ---- Kernel 4: GroupNorm(N=64) * g, then @ Wo^T ----------------------------
__global__ __launch_bounds__(128) void k_out(
    const float* __restrict__ Yb, const float* __restrict__ Gb,
    const float* __restrict__ ln_w, const float* __restrict__ ln_b,
    const float* __restrict__ Wo,
    float* __restrict__ Ob)
{
  __shared__ float sY[16 * LDA];
  __shared__ float sG[16 * LDA];
  __shared__ float smu[16], srs[16];
  const int row0 = blockIdx.x * 16;
  const int tid  = threadIdx.x;
  const int wave = tid >> 5;
  for (int i = tid; i < 16 * 64; i += 128) {
    const int r = i >> 6, c = i & 63;
    sY[r * LDA + c] = Yb[(size_t)(row0 + r) * 64 + c];
    sG[r * LDA + c] = Gb[(size_t)(row0 + r) * 64 + c];
  }
  __syncthreads();
  if (tid < 16) {
    float mu = 0.f, m2 = 0.f;
    for (int c = 0; c < 64; c++) {
      const float yv = sY[tid * LDA + c];
      mu += yv; m2 += yv * yv;
    }
    mu *= (1.f / 64.f); m2 *= (1.f / 64.f);
    smu[tid] = mu;
    srs[tid] = rsqrtf(m2 - mu * mu + LN_EPS);
  }
  __syncthreads();
  for (int i = tid; i < 16 * 64; i += 128) {
    const int r = i >> 6, c = i & 63;
    const float yn = (sY[r * LDA + c] - smu[r]) * srs[r] * ln_w[c] + ln_b[c];
    sY[r * LDA + c] = yn * sG[r * LDA + c];
  }
  __syncthreads();
  const int nt = wave;
  v8f acc = {};
  acc = wmma_ABt(sY, LDA, Wo + nt * 16 * 64, 64, 64, acc);
  const int l = tid & 31, n = l & 15, h = l >> 4;
#pragma unroll
  for (int v = 0; v < 8; v++)
    Ob[(size_t)(row0 + v + 8 * h) * 64 + nt * 16 + n] = acc[v];
}

// ---- Host launcher ---------------------------------------------------------
extern "C" void kernel_launch(void* const* d_in, const int* in_sizes, int n_in,
                              void* d_out, int out_size, void* d_ws, size_t ws_size,
                              hipStream_t stream)
{
  (void)in_sizes; (void)n_in; (void)out_size; (void)ws_size;
  const float* x       = (const float*)d_in[0];
  const float* tmx     = (const float*)d_in[1];
  const float* tmw     = (const float*)d_in[2];
  const float* tmk     = (const float*)d_in[3];
  const float* tmv     = (const float*)d_in[4];
  const float* tmr     = (const float*)d_in[5];
  const float* tmg     = (const float*)d_in[6];
  const float* maa_w1  = (const float*)d_in[7];
  const float* maa_w2  = (const float*)d_in[8];
  const float* t_decay = (const float*)d_in[9];
  const float* dec_w1  = (const float*)d_in[10];
  const float* dec_w2  = (const float*)d_in[11];
  const float* u       = (const float*)d_in[12];
  const float* Wr      = (const float*)d_in[13];
  const float* Wk      = (const float*)d_in[14];
  const float* Wv      = (const float*)d_in[15];
  const float* Wg      = (const float*)d_in[16];
  const float* Wo      = (const float*)d_in[17];
  const float* ln_w    = (const float*)d_in[18];
  const float* ln_b    = (const float*)d_in[19];

  const size_t S1 = (size_t)Mrows * Cc;      // 4,194,304 floats per buffer
  float* Rb = (float*)d_ws;
  float* Kb = Rb + S1;
  float* Vb = Kb + S1;
  float* Gb = Vb + S1;
  float* Hb = Gb + S1;                       // tanh(xw@dec_w1), then dec
  float* Yb = Hb + S1;
  float* Ob = (float*)d_out;

  const dim3 gTiles(Mrows / 16);             // 4096 tiles
  k_mix_proj<<<gTiles, 128, 0, stream>>>(x, tmx, tmw, tmk, tmv, tmr, tmg,
                                         maa_w1, maa_w2, dec_w1,
                                         Wr, Wk, Wv, Wg, Rb, Kb, Vb, Gb, Hb);
  k_decay<<<gTiles, 128, 0, stream>>>(dec_w2, t_decay, Hb);
  k_wkv<<<dim3(Bb), 64, 0, stream>>>(Rb, Kb, Vb, Hb, u, Yb);
  k_out<<<gTiles, 128, 0, stream>>>(Yb, Gb, ln_w, ln_b, Wo, Ob);
}